// QuantumLayer_37692632989746
// MI455X (gfx1250) — compile-verified
//
#include <hip/hip_runtime.h>
#include <math.h>

// ---------------------------------------------------------------------------
// 20-qubit statevector circuit for MI455X (gfx1250, wave32).
//  state: 2^20 complex64 (float2), 8 MB -> L2 resident.
//  Per layer: RX on wires 5..19 in LDS (256KB chunk butterflies),
//             RX on wires 0..4 as 64x64-real GEMM via V_WMMA_F32_16X16X4_F32,
//             CNOT ring = bit permutation, fused into next load as a gather.
// Wire w <-> flat index bit (19 - w).
// ---------------------------------------------------------------------------

typedef __attribute__((ext_vector_type(2))) float v2f;
typedef __attribute__((ext_vector_type(8))) float v8f;

#if __has_builtin(__builtin_amdgcn_wmma_f32_16x16x4_f32)
#define HAVE_WMMA_F32X4 1
#else
#define HAVE_WMMA_F32X4 0
#endif

// ---- inverse of the per-layer CNOT-ring permutation (gather index) --------
// y_i = prefix XOR of x over wires 0..i (i>=1); y_0 = XOR of wires 1..19.
// Inverse: x bit p (p<=17):  ybit_p ^ ybit_{p+1}
//          x bit 18       :  ybit18 ^ ybit19 ^ ybit0
//          x bit 19       :  ybit19 ^ ybit0
__device__ __forceinline__ unsigned inv_perm(unsigned y) {
  unsigned low = (y ^ (y >> 1)) & 0x3FFFFu;
  unsigned b19 = ((y >> 19) ^ y) & 1u;
  unsigned b18 = ((y >> 18) ^ (y >> 19) ^ y) & 1u;
  return low | (b18 << 18) | (b19 << 19);
}

// ---- prep: cos/sin(theta/2) table.  trig[0..19]=inputs, trig[20+l*20+w] ---
__global__ void prep_trig_kernel(const float* __restrict__ inp,
                                 const float* __restrict__ theta,
                                 float2* __restrict__ trig) {
  int tid = threadIdx.x;
  if (tid < 20) {
    float a = inp[tid] * 0.5f;
    trig[tid] = make_float2(cosf(a), sinf(a));
  } else if (tid < 140) {
    float a = theta[tid - 20] * 0.5f;
    trig[tid] = make_float2(cosf(a), sinf(a));
  }
}

// ---- prep: per-layer 64x64 real matrix for the 5 "global" RX gates --------
// U[j][k] = (-i)^popc(j^k) * prod_b (bit_b(j^k) ? sin_w : cos_w), w = 4-b.
// Real embedding (interleaved re/im): [[uR,-uI],[uI,uR]] blocks.
__global__ void prep_mats_kernel(const float2* __restrict__ trig,
                                 float* __restrict__ Mall) {
  int l = blockIdx.x;           // layer 0..5
  int tid = threadIdx.x;        // 1024 threads
  int j = tid >> 5, k = tid & 31;
  int x = j ^ k;
  float mag = 1.0f;
#pragma unroll
  for (int b = 0; b < 5; ++b) {
    int w = 4 - b;
    float2 cs = trig[20 + l * 20 + w];
    mag *= ((x >> b) & 1) ? cs.y : cs.x;
  }
  int pc = __popc((unsigned)x) & 3;
  float uR, uI;
  if (pc == 0)      { uR = mag;  uI = 0.0f; }
  else if (pc == 1) { uR = 0.0f; uI = -mag; }
  else if (pc == 2) { uR = -mag; uI = 0.0f; }
  else              { uR = 0.0f; uI = mag;  }
  float* Mm = Mall + l * 4096;
  int r0 = j << 1, c0 = k << 1;
  Mm[r0 * 64 + c0]           = uR;
  Mm[r0 * 64 + c0 + 1]       = -uI;
  Mm[(r0 + 1) * 64 + c0]     = uI;
  Mm[(r0 + 1) * 64 + c0 + 1] = uR;
}

// ---- RX on wires 5..19 inside a 2^15-amplitude LDS chunk ------------------
// mode 0: synthesize initial product state (input-angle RX on |0>).
// mode 1: gather-load through inverse CNOT-ring permutation (fused CNOT block).
__global__ void rx_local_kernel(const float2* __restrict__ in,
                                float2* __restrict__ out,
                                const float2* __restrict__ trig,
                                int layer, int mode) {
  extern __shared__ float2 sh[];          // 32768 float2 = 256 KB
  const int tid = threadIdx.x;            // 256 threads
  const unsigned chunkBase = (unsigned)blockIdx.x << 15;

  if (mode == 0) {
    // high 5 wires (block bits) factor
    float magH = 1.0f;
    int hb = blockIdx.x;
#pragma unroll
    for (int w = 0; w < 5; ++w) {
      float2 t = trig[w];
      magH *= ((hb >> (4 - w)) & 1) ? t.y : t.x;
    }
    int pcH = __popc((unsigned)hb);
    float cL[15], sL[15];
#pragma unroll
    for (int w = 5; w < 20; ++w) { float2 t = trig[w]; cL[w - 5] = t.x; sL[w - 5] = t.y; }
    for (int it = 0; it < 128; ++it) {
      int j = it * 256 + tid;
      float mag = magH;
#pragma unroll
      for (int w = 5; w < 20; ++w)
        mag *= ((j >> (19 - w)) & 1) ? sL[w - 5] : cL[w - 5];
      int pc = (pcH + __popc((unsigned)j)) & 3;
      float2 a;
      if (pc == 0)      a = make_float2(mag, 0.0f);
      else if (pc == 1) a = make_float2(0.0f, -mag);
      else if (pc == 2) a = make_float2(-mag, 0.0f);
      else              a = make_float2(0.0f, mag);
      sh[j] = a;
    }
  } else {
    for (int it = 0; it < 128; ++it) {
      int j = it * 256 + tid;
      unsigned y = chunkBase + (unsigned)j;
      sh[j] = in[inv_perm(y)];
    }
  }
  __syncthreads();

  const int tb = 20 + layer * 20;
  for (int b = 0; b < 15; ++b) {          // LDS bit b <-> wire 19-b
    float2 cs = trig[tb + (19 - b)];
    float c = cs.x, s = cs.y;
    for (int it = 0; it < 64; ++it) {
      int t = it * 256 + tid;             // pair id 0..16383
      int i0 = ((t >> b) << (b + 1)) | (t & ((1 << b) - 1));
      int i1 = i0 | (1 << b);
      float2 a0 = sh[i0], a1 = sh[i1];
      float2 n0 = make_float2(c * a0.x + s * a1.y, c * a0.y - s * a1.x);
      float2 n1 = make_float2(s * a0.y + c * a1.x, c * a1.y - s * a0.x);
      sh[i0] = n0; sh[i1] = n1;
    }
    __syncthreads();
  }

  for (int it = 0; it < 128; ++it) {
    int j = it * 256 + tid;
    out[chunkBase + j] = sh[j];
  }
}

#if HAVE_WMMA_F32X4
// ---- RX on wires 0..4: D(64 x 32768) = M(64x64) * S(64 x 32768) via WMMA --
// B/S column n = group (low 15 flat bits); row k = 2*q + (re/im), q = top-5.
// One wave owns one 16-group tile, all 4 row-blocks (64 WMMA ops).
__global__ void rx_global_wmma_kernel(const float* __restrict__ in,
                                      float* __restrict__ out,
                                      const float* __restrict__ M) {
  const int tid  = threadIdx.x;                 // 256 = 8 waves
  const int lane = tid & 31;
  const int gt   = (blockIdx.x << 3) | (tid >> 5);   // group tile 0..2047
  const int nloc = lane & 15;
  const int half = lane >> 4;                   // 0: K=0,1  1: K=2,3
  const int koff = half << 1;
  const unsigned n = ((unsigned)gt << 4) | (unsigned)nloc;   // group 0..32767
  const float2* in2 = (const float2*)in;
  float2* out2 = (float2*)out;

  v8f acc[4] = {};                              // rowblocks 0..3

  for (int kb = 0; kb < 16; ++kb) {
    const int kbase = kb << 2;
    // B fragment: rows kbase+koff, kbase+koff+1 at column n (k even -> re/im)
    const unsigned q = (unsigned)((kbase + koff) >> 1);
    float2 bv = in2[(q << 15) | n];
    v2f b; b.x = bv.x; b.y = bv.y;
#pragma unroll
    for (int rb = 0; rb < 4; ++rb) {
      const int row = (rb << 4) | nloc;
      v2f a;
      a.x = M[row * 64 + kbase + koff];
      a.y = M[row * 64 + kbase + koff + 1];
      acc[rb] = __builtin_amdgcn_wmma_f32_16x16x4_f32(
          false, a, false, b, (short)0, acc[rb], false, false);
    }
  }

  // Writeback: D rows (2q, 2q+1) are (re, im) of complex amp (q<<15)|n and
  // sit in adjacent accumulator VGPRs of the same lane -> one b64 store each.
#pragma unroll
  for (int rb = 0; rb < 4; ++rb) {
#pragma unroll
    for (int v = 0; v < 8; v += 2) {
      const int r = (rb << 4) | (v + (half << 3));   // even output row
      const unsigned q = (unsigned)(r >> 1);
      out2[(q << 15) | n] = make_float2(acc[rb][v], acc[rb][v + 1]);
    }
  }
}
#else
// ---- fallback: per-thread 32-amplitude register butterfly -----------------
__global__ void rx_global_scalar_kernel(const float2* __restrict__ in,
                                        float2* __restrict__ out,
                                        const float2* __restrict__ trig,
                                        int layer) {
  const unsigned g = blockIdx.x * 256u + threadIdx.x;   // 0..32767
  float2 v[32];
#pragma unroll
  for (int q = 0; q < 32; ++q) v[q] = in[((unsigned)q << 15) | g];
  const int tb = 20 + layer * 20;
#pragma unroll
  for (int b = 0; b < 5; ++b) {                 // group bit b <-> wire 4-b
    float2 cs = trig[tb + (4 - b)];
    float c = cs.x, s = cs.y;
#pragma unroll
    for (int t = 0; t < 16; ++t) {
      int i0 = ((t >> b) << (b + 1)) | (t & ((1 << b) - 1));
      int i1 = i0 | (1 << b);
      float2 a0 = v[i0], a1 = v[i1];
      v[i0] = make_float2(c * a0.x + s * a1.y, c * a0.y - s * a1.x);
      v[i1] = make_float2(s * a0.y + c * a1.x, c * a1.y - s * a0.x);
    }
  }
#pragma unroll
  for (int q = 0; q < 32; ++q) out[((unsigned)q << 15) | g] = v[q];
}
#endif

// ---- final: z_w = sum_x |amp(x)|^2 * (1 - 2*bit_w(P(x))), last perm fused -
__global__ void final_partial_kernel(const float2* __restrict__ st,
                                     float* __restrict__ partials) {
  __shared__ float red[256];
  const int tid = threadIdx.x;
  float acc[20];
#pragma unroll
  for (int w = 0; w < 20; ++w) acc[w] = 0.0f;

  const unsigned base = blockIdx.x * 4096u;
  for (int it = 0; it < 16; ++it) {
    unsigned x = base + it * 256u + tid;
    float2 a = st[x];
    float p = a.x * a.x + a.y * a.y;
    // forward CNOT-ring perm: t bit_p = XOR of x bits >= p (XOR suffix scan)
    unsigned t = x;
    t ^= t >> 1; t ^= t >> 2; t ^= t >> 4; t ^= t >> 8; t ^= t >> 16;
    unsigned y = (t & 0x7FFFFu) | ((((t ^ (x >> 19)) & 1u)) << 19);
#pragma unroll
    for (int w = 0; w < 20; ++w)
      acc[w] += ((y >> (19 - w)) & 1u) ? -p : p;
  }
  for (int w = 0; w < 20; ++w) {
    red[tid] = acc[w];
    __syncthreads();
    for (int sdist = 128; sdist > 0; sdist >>= 1) {
      if (tid < sdist) red[tid] += red[tid + sdist];
      __syncthreads();
    }
    if (tid == 0) partials[blockIdx.x * 20 + w] = red[0];
    __syncthreads();
  }
}

__global__ void final_reduce_kernel(const float* __restrict__ partials,
                                    float* __restrict__ out) {
  int tid = threadIdx.x;
  if (tid < 20) {
    float s = 0.0f;
    for (int b = 0; b < 256; ++b) s += partials[b * 20 + tid];
    out[tid] = s;
  }
}

// ---------------------------------------------------------------------------
extern "C" void kernel_launch(void* const* d_in, const int* in_sizes, int n_in,
                              void* d_out, int out_size, void* d_ws, size_t ws_size,
                              hipStream_t stream) {
  (void)in_sizes; (void)n_in; (void)out_size; (void)ws_size;
  const float* inp   = (const float*)d_in[0];   // 20 angles
  const float* theta = (const float*)d_in[1];   // 6*20 angles
  float* out = (float*)d_out;

  char* ws = (char*)d_ws;
  float2* A    = (float2*)ws;                               // 8 MB state
  float2* B    = (float2*)(ws + (size_t)(8u << 20));        // 8 MB state
  float2* trig = (float2*)(ws + (size_t)(16u << 20));       // 140 (c,s) pairs
  float*  Mall = (float*)(ws + (size_t)(16u << 20) + 4096); // 6 * 64*64 f32
  float*  part = (float*)(ws + (size_t)(16u << 20) + 4096 + 6 * 4096 * 4);

  (void)hipFuncSetAttribute((const void*)rx_local_kernel,
                            hipFuncAttributeMaxDynamicSharedMemorySize, 262144);

  prep_trig_kernel<<<1, 160, 0, stream>>>(inp, theta, trig);
#if HAVE_WMMA_F32X4
  prep_mats_kernel<<<6, 1024, 0, stream>>>(trig, Mall);
#endif

  // layer 0: synthesize initial product state + theta[0] wires 5..19
  rx_local_kernel<<<32, 256, 262144, stream>>>(B, A, trig, 0, 0);
#if HAVE_WMMA_F32X4
  rx_global_wmma_kernel<<<256, 256, 0, stream>>>((const float*)A, (float*)B,
                                                 Mall + 0 * 4096);
#else
  rx_global_scalar_kernel<<<128, 256, 0, stream>>>(A, B, trig, 0);
#endif

  for (int l = 1; l < 6; ++l) {
    // fused CNOT-ring perm of layer l-1 in the gather, then theta[l] wires 5..19
    rx_local_kernel<<<32, 256, 262144, stream>>>(B, A, trig, l, 1);
#if HAVE_WMMA_F32X4
    rx_global_wmma_kernel<<<256, 256, 0, stream>>>((const float*)A, (float*)B,
                                                   Mall + l * 4096);
#else
    rx_global_scalar_kernel<<<128, 256, 0, stream>>>(A, B, trig, l);
#endif
  }

  // last layer's CNOT perm fused into expectation; deterministic 2-stage sum
  final_partial_kernel<<<256, 256, 0, stream>>>(B, part);
  final_reduce_kernel<<<1, 32, 0, stream>>>(part, out);
}